// FieldTransformer_35682588295622
// MI455X (gfx1250) — compile-verified
//
#include <hip/hip_runtime.h>

typedef __attribute__((ext_vector_type(2))) float v2f;
typedef __attribute__((ext_vector_type(8))) float v8f;

#define NQ    8192
#define NPTS  32768
#define CH    128
#define KNN   64
#define QT    16           // queries per workgroup (one WMMA M-tile)
#define NT    (NPTS / 16)  // 2048 N-tiles
#define NWAVE 8
#define TPB   256
#define CCAP  96           // per-query candidate capacity (prefix-tie spill list)

// Pack neural_pos [N,3] -> (x, y, z, |p|^2) float4 for aligned float2 B staging.
__global__ void ft_pack_pos(const float* __restrict__ pos, float4* __restrict__ out) {
    int i = blockIdx.x * blockDim.x + threadIdx.x;
    if (i < NPTS) {
        float x = pos[3 * i + 0], y = pos[3 * i + 1], z = pos[3 * i + 2];
        out[i] = make_float4(x, y, z, x * x + y * y + z * z);
    }
}

__global__ __launch_bounds__(TPB) void ft_knn_kernel(
    const float* __restrict__ qpos,
    const float* __restrict__ feat,
    const float2* __restrict__ pos2,   // pos4 viewed as pairs: [2n]=(x,y), [2n+1]=(z,|p|^2)
    float* __restrict__ out)
{
    __shared__ unsigned int hist[QT][256];      // 16 KB radix histograms
    __shared__ unsigned int prefix[QT];
    __shared__ unsigned int need[QT];
    __shared__ unsigned int cnt[QT];
    __shared__ unsigned int ccnt[QT];
    __shared__ float        qn2[QT];
    __shared__ unsigned int cand_key[QT][CCAP]; // 6 KB
    __shared__ int          cand_idx[QT][CCAP]; // 6 KB
    __shared__ int          a_idx[QT][KNN];     // accepted neighbor indices
    __shared__ float        a_w[QT][KNN];       // accepted weights

    const int tid   = threadIdx.x;
    const int lane  = tid & 31;
    const int wave  = tid >> 5;
    const int qbase = blockIdx.x * QT;
    const int nlo   = lane & 15;   // N (or M-row for A) within tile
    const int hi    = lane >> 4;   // lane half
    const int hi8   = hi << 3;
    // this lane's fixed float2 slot within a 16-point tile (32 float2 per tile)
    const int bslot = nlo * 2 + hi;

    if (tid < QT) {
        float x = qpos[3 * (qbase + tid) + 0];
        float y = qpos[3 * (qbase + tid) + 1];
        float z = qpos[3 * (qbase + tid) + 2];
        qn2[tid]    = x * x + y * y + z * z;
        prefix[tid] = 0u;
        need[tid]   = KNN;
    }
    __syncthreads();

    // A matrix (16x4 f32), row M = nlo.
    // lanes 0-15 hold K0,K1 = (-2qx, -2qy); lanes 16-31 hold K2,K3 = (-2qz, 1).
    float ax = qpos[3 * (qbase + nlo) + 0];
    float ay = qpos[3 * (qbase + nlo) + 1];
    float az = qpos[3 * (qbase + nlo) + 2];
    v2f A;
    if (hi == 0) { A.x = -2.0f * ax; A.y = -2.0f * ay; }
    else         { A.x = -2.0f * az; A.y = 1.0f; }

    // C init: c[r] = |q_M|^2 with M = r + 8*hi  ->  D = |q|^2 + |p|^2 - 2 q.p = d^2
    v8f cinit;
    #pragma unroll
    for (int r = 0; r < 8; ++r) cinit[r] = qn2[r + hi8];

    // One 16x16 tile of squared distances via f32 WMMA.
    // B matrix (4x16 f32): lanes 0-15 hold K0,K1 = (px,py); lanes 16-31 hold K2,K3 = (pz,|p|^2).
    // Each lane's float2 load IS its two B registers - no selects needed.
    auto tile_d2 = [&](float2 p) -> v8f {
        v2f B; B.x = p.x; B.y = p.y;
        return __builtin_amdgcn_wmma_f32_16x16x4_f32(
            false, A, false, B, (short)0, cinit, false, false);
    };

    // ---------------- 2-pass radix select (top 16 key bits) ----------------
    for (int pass = 0; pass < 2; ++pass) {
        for (int i = tid; i < QT * 256; i += TPB) ((unsigned int*)hist)[i] = 0u;
        __syncthreads();

        unsigned int pfxr[8];
        #pragma unroll
        for (int r = 0; r < 8; ++r) pfxr[r] = prefix[r + hi8];

        const int shift = 24 - 8 * pass;
        float2 p = pos2[wave * 32 + bslot];
        for (int t = wave; t < NT; t += NWAVE) {
            const int tn = t + NWAVE;
            float2 pn = pos2[(tn < NT ? tn : t) * 32 + bslot];   // prefetch next tile
            v8f dm = tile_d2(p);
            #pragma unroll
            for (int r = 0; r < 8; ++r) {
                float d2 = dm[r] > 0.0f ? dm[r] : 0.0f;
                unsigned int key = __float_as_uint(d2);
                bool ok = (pass == 0) ? true : ((key >> 24) == pfxr[r]);
                if (ok) atomicAdd(&hist[r + hi8][(key >> shift) & 255u], 1u);
            }
            p = pn;
        }
        __syncthreads();

        if (tid < QT) {
            unsigned int nd = need[tid];
            unsigned int cum = 0;
            int dsel = 255;
            for (int b = 0; b < 256; ++b) {
                unsigned int cb = hist[tid][b];
                if (cum + cb >= nd) { dsel = b; break; }
                cum += cb;
            }
            prefix[tid] = (prefix[tid] << 8) | (unsigned int)dsel;
            need[tid]   = nd - cum;
        }
        __syncthreads();
    }

    // ---------------- final sweep: direct accept + tie-candidate spill ----------------
    unsigned int pfx16r[8];
    #pragma unroll
    for (int r = 0; r < 8; ++r) pfx16r[r] = prefix[r + hi8];
    if (tid < QT) { cnt[tid] = 0u; ccnt[tid] = 0u; }
    for (int i = tid; i < QT * KNN; i += TPB) { ((int*)a_idx)[i] = 0; ((float*)a_w)[i] = 0.0f; }
    __syncthreads();

    {
        float2 p = pos2[wave * 32 + bslot];
        for (int t = wave; t < NT; t += NWAVE) {
            const int tn = t + NWAVE;
            float2 pn = pos2[(tn < NT ? tn : t) * 32 + bslot];
            v8f dm = tile_d2(p);
            const int n = t * 16 + nlo;
            #pragma unroll
            for (int r = 0; r < 8; ++r) {
                const int m = r + hi8;
                float d2 = dm[r] > 0.0f ? dm[r] : 0.0f;
                unsigned int key = __float_as_uint(d2);
                unsigned int h16 = key >> 16;
                if (h16 < pfx16r[r]) {
                    // strictly inside the top-64 (exact count from histograms)
                    unsigned int pi = atomicAdd(&cnt[m], 1u);
                    if (pi < KNN) {
                        float dist = sqrtf(fmaxf(d2, 1e-12f));
                        a_idx[m][pi] = n;
                        a_w[m][pi]   = 1.0f / (dist + 1e-5f);
                    }
                } else if (h16 == pfx16r[r]) {
                    // shares the kth neighbor's 16-bit prefix: resolve exactly later
                    unsigned int ci = atomicAdd(&ccnt[m], 1u);
                    if (ci < CCAP) { cand_key[m][ci] = key; cand_idx[m][ci] = n; }
                }
            }
            p = pn;
        }
    }
    __syncthreads();

    // exact resolution of prefix-equal candidates (typically 1-3 entries/query)
    if (tid < QT) {
        const int m    = tid;
        const int base = (int)cnt[m];
        int nc   = (int)ccnt[m]; if (nc > CCAP) nc = CCAP;
        int want = KNN - base;   if (want > nc) want = nc;
        for (int s = 0; s < want; ++s) {
            unsigned int bk = 0xFFFFFFFFu; int bi = -1;
            for (int j = 0; j < nc; ++j) {
                unsigned int k2 = cand_key[m][j];
                if (k2 < bk) { bk = k2; bi = j; }
            }
            if (bi >= 0) {
                cand_key[m][bi] = 0xFFFFFFFFu;   // mark used
                float d2   = __uint_as_float(bk);
                float dist = sqrtf(fmaxf(d2, 1e-12f));
                a_idx[m][base + s] = cand_idx[m][bi];
                a_w[m][base + s]   = 1.0f / (dist + 1e-5f);
            }
        }
    }
    __syncthreads();

    // ---------------- gather + weighted sum (L2-resident features) ----------------
    const int c  = tid & (CH - 1);
    const int mh = tid >> 7;          // two queries processed per sweep
    for (int mm = 0; mm < QT; mm += 2) {
        int m = mm + mh;
        float acc = 0.0f;
        #pragma unroll 4
        for (int j = 0; j < KNN; ++j) {
            int   idx = a_idx[m][j];
            float w   = a_w[m][j];
            acc += w * feat[idx * CH + c];
        }
        out[(qbase + m) * CH + c] = acc;
    }
}

extern "C" void kernel_launch(void* const* d_in, const int* in_sizes, int n_in,
                              void* d_out, int out_size, void* d_ws, size_t ws_size,
                              hipStream_t stream) {
    const float* qpos = (const float*)d_in[0];   // [8192,3]
    const float* feat = (const float*)d_in[1];   // [32768,128]
    const float* npos = (const float*)d_in[2];   // [32768,3]
    (void)in_sizes; (void)n_in; (void)out_size; (void)ws_size;

    float4* pos4 = (float4*)d_ws;                // 32768 * 16 B = 512 KB scratch
    ft_pack_pos<<<(NPTS + 255) / 256, 256, 0, stream>>>(npos, pos4);
    ft_knn_kernel<<<NQ / QT, TPB, 0, stream>>>(qpos, feat, (const float2*)pos4,
                                               (float*)d_out);
}